// GCRNN_52304111731110
// MI455X (gfx1250) — compile-verified
//
#include <hip/hip_runtime.h>
#include <hip/hip_bf16.h>
#include <math.h>

#define Bn 32
#define Tn 128
#define Nn 128
#define Dn 64
#define Qn 5
#define En 8
#define Pn 768
#define CIN 129      // 2*D+1
#define YS  132      // padded row stride for Y / combined / adj (132 % 64 == 4 -> conflict-free)
#define WSTRD 68     // W stage row stride (272B pitch: 16B-aligned, half-wave bank offset = 8)
#define ALPHAc 0.5f

typedef float v2f __attribute__((ext_vector_type(2)));
typedef float v8f __attribute__((ext_vector_type(8)));

// ---- LDS float offsets (dynamic LDS, ~247KB; CDNA5 WGP has 320KB) ----
#define OFF_Y    0
#define OFF_CMB  (Nn*YS)                 // 16896
#define OFF_ADJ  (2*Nn*YS)               // 33792
#define OFF_WST  (3*Nn*YS)               // 50688
#define OFF_RAR  (OFF_WST + YS*WSTRD)    // +8976 = 59664
#define OFF_MSK  (OFF_RAR + Nn)
#define OFF_VTO  (OFF_MSK + Nn)
#define OFF_VV   (OFF_VTO + Nn)
#define OFF_BIAS (OFF_VV + Qn*Nn)
#define LDS_FLOATS (OFF_BIAS + 3*Qn*Dn)  // 61648 floats = 246592 bytes

__device__ __forceinline__ v8f wmma4(v2f a, v2f b, v8f c) {
  // V_WMMA_F32_16X16X4_F32 : D = A(16x4) * B(4x16) + C, fp32 in/out
  return __builtin_amdgcn_wmma_f32_16x16x4_f32(false, a, false, b, (short)0, c,
                                               false, false);
}

// ---------------------------------------------------------------------------
// Kernel A: batch-independent precompute of var_vector (N x Q) and the
// softmax adjacency (N x N). One block, one thread per variable n.
// ---------------------------------------------------------------------------
__global__ void __launch_bounds__(128)
gcrnn_precompute(const float* __restrict__ plm,   // (N,P)
                 const float* __restrict__ Ws1, const float* __restrict__ bs1,
                 const float* __restrict__ Ws2, const float* __restrict__ bs2,
                 const float* __restrict__ Wg1, const float* __restrict__ bg1,
                 const float* __restrict__ Wg2, const float* __restrict__ bg2,
                 float* __restrict__ vv_g,        // (N,Q)
                 float* __restrict__ adj_g)       // (N,N)
{
  __shared__ float sg[Nn][En];
  const int n = threadIdx.x;

  float vv[Qn];
  for (int q = 0; q < Qn; ++q) vv[q] = bs2[q];
  float ge[En];
  for (int e = 0; e < En; ++e) ge[e] = bg2[e];

  for (int j = 0; j < 2 * Dn; ++j) {
    float s1 = bs1[j];
    float s2 = bg1[j];
    for (int p = 0; p < Pn; ++p) {
      float x = plm[n * Pn + p];
      s1 = fmaf(x, Ws1[p * (2 * Dn) + j], s1);
      s2 = fmaf(x, Wg1[p * (2 * Dn) + j], s2);
    }
    s1 = fmaxf(s1, 0.f);
    s2 = fmaxf(s2, 0.f);
    for (int q = 0; q < Qn; ++q) vv[q] = fmaf(s1, Ws2[j * Qn + q], vv[q]);
    for (int e = 0; e < En; ++e) ge[e] = fmaf(s2, Wg2[j * En + e], ge[e]);
  }
  for (int q = 0; q < Qn; ++q) vv_g[n * Qn + q] = vv[q];

  float nrm = 0.f;
  for (int e = 0; e < En; ++e) nrm += ge[e] * ge[e];
  nrm = fmaxf(sqrtf(nrm), 1e-12f);
  for (int e = 0; e < En; ++e) sg[n][e] = ge[e] / nrm;
  __syncthreads();

  float sc[Nn];
  float mx = -3.4e38f;
  for (int m = 0; m < Nn; ++m) {
    float d = 0.f;
    for (int e = 0; e < En; ++e) d += sg[n][e] * sg[m][e];
    sc[m] = d;
    mx = fmaxf(mx, d);
  }
  float sum = 0.f;
  for (int m = 0; m < Nn; ++m) { sc[m] = expf(sc[m] - mx); sum += sc[m]; }
  float inv = 1.f / sum;
  for (int m = 0; m < Nn; ++m) adj_g[n * Nn + m] = sc[m] * inv;
}

// ---------------------------------------------------------------------------
// Helpers for the main kernel
// ---------------------------------------------------------------------------
// Stage one (CIN x 64) gate-weight matrix into LDS, permuted as [k][l16][ot]
// so that one b128 read per k-row serves all four 16-wide output tiles.
__device__ __forceinline__ void stage_W(float* __restrict__ WST,
                                        const float* __restrict__ Wsrc,  // (CIN,Dn)
                                        int tid) {
#pragma unroll
  for (int i = 0; i < 33; ++i) {               // 132*64 / 256 = 33
    int e = tid + 256 * i;
    int rowk = e >> 6, c = e & 63;
    float v = (rowk < CIN) ? Wsrc[rowk * Dn + c] : 0.f;
    WST[rowk * WSTRD + ((c & 15) << 2) + (c >> 4)] = v;
  }
}

// acc(16-row n-tile x 64 cols) += diag(vv[:,q]) * Amat(128x132) @ WST(132x64)
// K outer / o-tiles inner: per K-step 1 A-pair + 2 b128 B-reads -> 4 WMMAs.
__device__ __forceinline__ void mm_accum(v8f (&acc)[4],
                                         const float* __restrict__ Amat,
                                         const float* __restrict__ WST,
                                         const float* __restrict__ VV,
                                         int q, int rowA, int hlf, int l16) {
  float vvq = VV[q * Nn + rowA];
  const float* Arow = Amat + rowA * YS;
  const int boff = l16 << 2;
  for (int kt = 0; kt < 33; ++kt) {            // K = 132
    int kb = kt * 4 + hlf * 2;
    v2f av; av.x = Arow[kb] * vvq;
            av.y = Arow[kb + 1] * vvq;
    float4 b0 = *(const float4*)&WST[kb * WSTRD + boff];        // row kb, ot 0..3
    float4 b1 = *(const float4*)&WST[(kb + 1) * WSTRD + boff];  // row kb+1
    v2f bv;
    bv.x = b0.x; bv.y = b1.x; acc[0] = wmma4(av, bv, acc[0]);
    bv.x = b0.y; bv.y = b1.y; acc[1] = wmma4(av, bv, acc[1]);
    bv.x = b0.z; bv.y = b1.z; acc[2] = wmma4(av, bv, acc[2]);
    bv.x = b0.w; bv.y = b1.w; acc[3] = wmma4(av, bv, acc[3]);
  }
}

// ---------------------------------------------------------------------------
// Kernel B: one workgroup (256 threads = 8 waves) per batch element.
// Whole recurrence state lives in LDS; all GEMMs on v_wmma_f32_16x16x4_f32.
// ---------------------------------------------------------------------------
__global__ void __launch_bounds__(256)
gcrnn_main(const float* __restrict__ obs,    // (B,T,N,D)
           const float* __restrict__ msk,    // (B,T,N)
           const int*   __restrict__ lens,   // (B,1)
           const float* __restrict__ avg,    // (B,T,N)
           const float* __restrict__ rarW,   // (N,N)
           const float* __restrict__ W_r, const float* __restrict__ W_u,
           const float* __restrict__ W_c,    // each (Q,CIN,D)
           const float* __restrict__ b_r, const float* __restrict__ b_u,
           const float* __restrict__ b_c,    // each (Q,D)
           const float* __restrict__ vv_g,   // (N,Q)
           const float* __restrict__ adj_g,  // (N,N)
           float* __restrict__ out)          // (B,N,D)
{
  extern __shared__ float L[];
  float* Y    = L + OFF_Y;     // rows n, cols: [0..63]=x, 64=rarity, [65..128]=h, [129..131]=0
  float* CMB  = L + OFF_CMB;   // combined (128 x 132)
  float* ADJ  = L + OFF_ADJ;   // cur_adj  (128 x 132, cols 0..127 used)
  float* WST  = L + OFF_WST;   // staged gate weight, permuted [k][l16][ot], stride 68
  float* RAR  = L + OFF_RAR;
  float* MSK  = L + OFF_MSK;
  float* VTO  = L + OFF_VTO;
  float* VV   = L + OFF_VV;    // [q][n]
  float* BIAS = L + OFF_BIAS;  // [gate r/u/c][q][o]

  const int tid  = threadIdx.x;
  const int b    = blockIdx.x;
  const int lane = tid & 31;
  const int w    = tid >> 5;        // wave id = n-tile id (8 tiles of 16 rows)
  const int hlf  = lane >> 4;       // half-wave (K split for wmma f32 16x16x4)
  const int l16  = lane & 15;
  const int rowA = w * 16 + l16;    // A-tile row this lane services

  // ---- one-time block init ----
  for (int e = tid; e < Nn * YS; e += 256) Y[e] = 0.f;   // zeros h + pad cols
  if (tid < Nn) {
    float s = 0.f;
    for (int t = 0; t < Tn; ++t) s += msk[((size_t)b * Tn + t) * Nn + tid];
    VTO[tid] = s;                                        // var_total_obs[b,n]
  }
  for (int e = tid; e < Qn * Nn; e += 256) {
    int q = e / Nn, n = e % Nn;
    VV[e] = vv_g[n * Qn + q];
  }
  for (int e = tid; e < 3 * Qn * Dn; e += 256) {
    int g = e / (Qn * Dn), r = e % (Qn * Dn);
    const float* bp = (g == 0) ? b_r : ((g == 1) ? b_u : b_c);
    BIAS[e] = bp[r];
  }
  const int len = lens[b];
  __syncthreads();

  for (int t = 0; t < Tn; ++t) {
    // ---- phase 1: rarity, mask, x into Y ----
    if (tid < Nn) {
      int n = tid;
      size_t ix = ((size_t)b * Tn + t) * Nn + n;
      float m = msk[ix];
      MSK[n] = m;
      float rr = ALPHAc * tanhf(avg[ix] / (VTO[n] + 1.f));
      RAR[n] = rr;
      Y[n * YS + 64] = rr;
    }
    {
      const float4* src = (const float4*)(obs + ((size_t)b * Tn + t) * Nn * Dn);
#pragma unroll
      for (int i = 0; i < 8; ++i) {                 // 2048 float4 = 128x64 x
        int f = tid + 256 * i;
        int base = f * 4, row = base >> 6, col = base & 63;
        float4 v = src[f];
        float* dst = &Y[row * YS + col];
        dst[0] = v.x; dst[1] = v.y; dst[2] = v.z; dst[3] = v.w;
      }
    }
    __syncthreads();

    // ---- phase 2: cur_adj = adj*(1+rsm)*maskp*(1-I) + I ----
    for (int i = 0; i < 64; ++i) {
      int e = tid + 256 * i;
      int n = e >> 7, m_ = e & 127;
      float base = adj_g[e];
      float rs = -rarW[e] * fabsf(RAR[n] - RAR[m_]);
      float mp = MSK[n] * MSK[m_];
      ADJ[n * YS + m_] = (n == m_) ? 1.f : base * (1.f + rs) * mp;
    }
    __syncthreads();

    // ---- phase 3: combined = cur_adj(128x128) @ Y(128x[132..144]) ----
    // K outer, 9 column-tiles inner: A-pair loaded once per K-step; the
    // zero-select exists only for the partial 9th tile.
    {
      v8f acc[9];
#pragma unroll
      for (int ct = 0; ct < 9; ++ct)
        acc[ct] = (v8f){0.f, 0.f, 0.f, 0.f, 0.f, 0.f, 0.f, 0.f};
      const float* Arow = ADJ + rowA * YS;
      for (int kt = 0; kt < 32; ++kt) {             // K = 128
        int kb = kt * 4 + hlf * 2;
        v2f av; av.x = Arow[kb];
                av.y = Arow[kb + 1];
        const float* Yk0 = Y + kb * YS;
        const float* Yk1 = Yk0 + YS;
#pragma unroll
        for (int ct = 0; ct < 9; ++ct) {
          int col = ct * 16 + l16;
          v2f bv;
          if (ct < 8) {
            bv.x = Yk0[col];
            bv.y = Yk1[col];
          } else {
            int cv = (col < YS) ? 1 : 0;
            int colc = cv ? col : 0;
            float b0 = Yk0[colc];
            float b1 = Yk1[colc];
            bv.x = cv ? b0 : 0.f;
            bv.y = cv ? b1 : 0.f;
          }
          acc[ct] = wmma4(av, bv, acc[ct]);
        }
      }
#pragma unroll
      for (int ct = 0; ct < 9; ++ct) {
        int col = ct * 16 + l16;
#pragma unroll
        for (int v = 0; v < 8; ++v) {
          int row = w * 16 + v + hlf * 8;
          if (col < YS) CMB[row * YS + col] = acc[ct][v];   // pad cols get exact 0
        }
      }
    }
    __syncthreads();

    // ---- phase 4: gates. gate = sigma( sum_q diag(vv_q)*In @ W[q] + bias ) ----
    v8f accR[4] = {{0,0,0,0,0,0,0,0},{0,0,0,0,0,0,0,0},{0,0,0,0,0,0,0,0},{0,0,0,0,0,0,0,0}};
    v8f accU[4] = {{0,0,0,0,0,0,0,0},{0,0,0,0,0,0,0,0},{0,0,0,0,0,0,0,0},{0,0,0,0,0,0,0,0}};
    v8f accC[4] = {{0,0,0,0,0,0,0,0},{0,0,0,0,0,0,0,0},{0,0,0,0,0,0,0,0},{0,0,0,0,0,0,0,0}};

#pragma unroll 1
    for (int q = 0; q < Qn; ++q) {                  // r gate (input = combined)
      stage_W(WST, W_r + (size_t)q * CIN * Dn, tid);
      if (q + 1 < Qn) __builtin_prefetch(W_r + (size_t)(q + 1) * CIN * Dn, 0, 0);
      __syncthreads();
      mm_accum(accR, CMB, WST, VV, q, rowA, hlf, l16);
      __syncthreads();
    }
#pragma unroll 1
    for (int q = 0; q < Qn; ++q) {                  // u gate (input = combined)
      stage_W(WST, W_u + (size_t)q * CIN * Dn, tid);
      if (q + 1 < Qn) __builtin_prefetch(W_u + (size_t)(q + 1) * CIN * Dn, 0, 0);
      __syncthreads();
      mm_accum(accU, CMB, WST, VV, q, rowA, hlf, l16);
      __syncthreads();
    }

    // apply r: h_r = mask ? r*h : h  (in place in Y's h columns)
#pragma unroll
    for (int ot = 0; ot < 4; ++ot) {
      int o = ot * 16 + l16;
#pragma unroll
      for (int v = 0; v < 8; ++v) {
        int row = w * 16 + v + hlf * 8;
        float bias = 0.f;
        for (int q = 0; q < Qn; ++q)
          bias = fmaf(VV[q * Nn + row], BIAS[(0 * Qn + q) * Dn + o], bias);
        float r = 1.f / (1.f + expf(-(accR[ot][v] + bias)));
        float h = Y[row * YS + 65 + o];
        Y[row * YS + 65 + o] = (MSK[row] > 0.f) ? r * h : h;
      }
    }
    __syncthreads();

#pragma unroll 1
    for (int q = 0; q < Qn; ++q) {                  // cand gate (input = [x|rar|h_r] = Y)
      stage_W(WST, W_c + (size_t)q * CIN * Dn, tid);
      if (q + 1 < Qn) __builtin_prefetch(W_c + (size_t)(q + 1) * CIN * Dn, 0, 0);
      __syncthreads();
      mm_accum(accC, Y, WST, VV, q, rowA, hlf, l16);
      __syncthreads();
    }

    // ---- final combine: h_new, out ----
    const bool isLast = (t == len - 1);
#pragma unroll
    for (int ot = 0; ot < 4; ++ot) {
      int o = ot * 16 + l16;
#pragma unroll
      for (int v = 0; v < 8; ++v) {
        int row = w * 16 + v + hlf * 8;
        float bU = 0.f, bC = 0.f;
        for (int q = 0; q < Qn; ++q) {
          float vvq = VV[q * Nn + row];
          bU = fmaf(vvq, BIAS[(1 * Qn + q) * Dn + o], bU);
          bC = fmaf(vvq, BIAS[(2 * Qn + q) * Dn + o], bC);
        }
        float u  = 1.f / (1.f + expf(-(accU[ot][v] + bU)));
        float c  = tanhf(accC[ot][v] + bC);
        float hr = Y[row * YS + 65 + o];
        float hn = (MSK[row] > 0.f) ? (1.f - u) * hr + u * c : hr;
        Y[row * YS + 65 + o] = hn;
        if (isLast) out[((size_t)b * Nn + row) * Dn + o] = hn;
      }
    }
    __syncthreads();
  }
}

// ---------------------------------------------------------------------------
extern "C" void kernel_launch(void* const* d_in, const int* in_sizes, int n_in,
                              void* d_out, int out_size, void* d_ws, size_t ws_size,
                              hipStream_t stream) {
  const float* obs  = (const float*)d_in[0];
  const float* msk  = (const float*)d_in[1];
  const int*   lens = (const int*)  d_in[2];
  const float* avg  = (const float*)d_in[3];
  const float* plm  = (const float*)d_in[4];
  const float* rarW = (const float*)d_in[5];
  const float* W_u  = (const float*)d_in[6];
  const float* b_u  = (const float*)d_in[7];
  const float* W_r  = (const float*)d_in[8];
  const float* b_r  = (const float*)d_in[9];
  const float* W_c  = (const float*)d_in[10];
  const float* b_c  = (const float*)d_in[11];
  const float* Ws1  = (const float*)d_in[12];
  const float* bs1  = (const float*)d_in[13];
  const float* Ws2  = (const float*)d_in[14];
  const float* bs2  = (const float*)d_in[15];
  const float* Wg1  = (const float*)d_in[16];
  const float* bg1  = (const float*)d_in[17];
  const float* Wg2  = (const float*)d_in[18];
  const float* bg2  = (const float*)d_in[19];
  float* out = (float*)d_out;

  float* vv_g  = (float*)d_ws;         // 128*5
  float* adj_g = vv_g + 1024;          // 128*128

  gcrnn_precompute<<<1, 128, 0, stream>>>(plm, Ws1, bs1, Ws2, bs2,
                                          Wg1, bg1, Wg2, bg2, vv_g, adj_g);

  size_t lds_bytes = (size_t)LDS_FLOATS * sizeof(float);   // ~247 KB (<320 KB/WGP)
  gcrnn_main<<<dim3(Bn), dim3(256), lds_bytes, stream>>>(
      obs, msk, lens, avg, rarW, W_r, W_u, W_c, b_r, b_u, b_c,
      vv_g, adj_g, out);
}